// _MultiheadAttention_57440892617323
// MI455X (gfx1250) — compile-verified
//
#include <hip/hip_runtime.h>
#include <hip/hip_bf16.h>
#include <math.h>

// ---------------------------------------------------------------------------
// Re-Attention (DeepViT) forward for MI455X / gfx1250, fp32 via
// V_WMMA_F32_16X16X4_F32 (wave32) + GLOBAL_LOAD_ASYNC_TO_LDS_B128 staging.
//
// Shapes: b=8, n=1024, d_model=512, heads=8, head_dim=64
// d_out = [ out (8*1024*512) | attn (8*8*1024*1024) ]   (fp32, concatenated)
// d_ws  = [ qkv (8192*1536)  | O (8*1024*512) ]         (fp32)
// ---------------------------------------------------------------------------

typedef __attribute__((ext_vector_type(2))) float v2f;
typedef __attribute__((ext_vector_type(8))) float v8f;

static __device__ __forceinline__ v8f wmma4(v2f a, v2f b, v8f c) {
  // D = A(16x4) * B(4x16) + C(16x16), fp32 in/out
  return __builtin_amdgcn_wmma_f32_16x16x4_f32(
      /*neg_a=*/false, a, /*neg_b=*/false, b,
      /*c_mod=*/(short)0, c, /*reuse_a=*/false, /*reuse_b=*/false);
}

// Async DMA: 16 bytes global -> LDS, tracked by ASYNCcnt (no VGPR staging).
static __device__ __forceinline__ void async_ld_lds_b128(void* lds, const void* gp) {
  const unsigned int lo = (unsigned int)(unsigned long long)lds; // LDS byte offset
  const unsigned long long ga = (unsigned long long)gp;
  asm volatile("global_load_async_to_lds_b128 %0, %1, off"
               :: "v"(lo), "v"(ga) : "memory");
}
static __device__ __forceinline__ void wait_async0() {
  asm volatile("s_wait_asynccnt 0x0" ::: "memory");
}

// ---------------------------------------------------------------------------
// Generic C[M,N] = A[M,K] @ B[K,N] + bias[N]
// block = 256 threads (8 waves), each wave one 16x16 tile (shared N-tile).
// B tile staged in LDS via async DMA in 64-deep K chunks (4 KB).
// grid.x = N/16, grid.y = M/128
// ---------------------------------------------------------------------------
__global__ __launch_bounds__(256) void wmma_gemm_bias(
    const float* __restrict__ A, const float* __restrict__ B,
    const float* __restrict__ bias, float* __restrict__ C,
    int K, int lda, int ldb, int ldc)
{
  __shared__ float Bs[64 * 16];    // [k within chunk][n]

  const int lane = threadIdx.x & 31;
  const int wave = threadIdx.x >> 5;
  const int half = lane >> 4;      // 0: lanes 0-15, 1: lanes 16-31
  const int l16  = lane & 15;

  const int row0    = (blockIdx.y * 8 + wave) * 16;
  const int colBase = blockIdx.x * 16;
  const int col     = colBase + l16;

  const float* __restrict__ Arow = A + (size_t)(row0 + l16) * lda;

  // async-stage descriptor: one b128 per thread per chunk
  const int skk = threadIdx.x >> 2;        // 0..63 (k within chunk)
  const int scc = (threadIdx.x & 3) * 4;   // 0,4,8,12 (n quad)

  v8f acc = {};
  for (int kc = 0; kc < K; kc += 64) {
    __syncthreads();   // previous chunk fully consumed
    async_ld_lds_b128(&Bs[skk * 16 + scc],
                      B + (size_t)(kc + skk) * ldb + colBase + scc);
    wait_async0();
    __syncthreads();   // all waves' DMA landed

#pragma unroll
    for (int k4 = 0; k4 < 64; k4 += 4) {
      v2f a = *(const v2f*)(Arow + kc + k4 + 2 * half);
      v2f b;
      b.x = Bs[(k4 + 2 * half)     * 16 + l16];
      b.y = Bs[(k4 + 2 * half + 1) * 16 + l16];
      acc = wmma4(a, b, acc);
    }
  }

  const float bv = bias[col];
#pragma unroll
  for (int i = 0; i < 8; ++i) {
    // C/D layout: VGPR i -> row (half*8 + i), col = l16
    C[(size_t)(row0 + half * 8 + i) * ldc + col] = acc[i] + bv;
  }
}

// ---------------------------------------------------------------------------
// scores = softmax( (Q @ K^T) * 0.125 )   per (b, h), streamed to attn buffer
// block = 256 threads (8 waves); one block per (b*8+h, i-tile of 16 rows).
// Wave w owns columns [w*128, w*128+128); full 16x1024 panel staged in LDS.
// ---------------------------------------------------------------------------
__global__ __launch_bounds__(256) void attn_scores_softmax(
    const float* __restrict__ qkv, float* __restrict__ attn)
{
  __shared__ float S[16 * 1024];   // 64 KB score panel

  const int lane = threadIdx.x & 31;
  const int wave = threadIdx.x >> 5;
  const int half = lane >> 4;
  const int l16  = lane & 15;

  const int bh = blockIdx.x;           // b*8 + h
  const int b  = bh >> 3;
  const int h  = bh & 7;
  const int i0 = blockIdx.y * 16;      // row tile

  const float* __restrict__ Q  = qkv + (size_t)(b * 1024) * 1536 + h * 64;
  const float* __restrict__ Kp = qkv + (size_t)(b * 1024) * 1536 + 512 + h * 64;

  const int j0w = wave * 128;
  const float* __restrict__ Qrow = Q + (size_t)(i0 + l16) * 1536;

  v8f acc[8] = {};
  for (int d = 0; d < 64; d += 4) {
    v2f a = *(const v2f*)(Qrow + d + 2 * half);
#pragma unroll
    for (int t = 0; t < 8; ++t) {
      const int j = j0w + t * 16 + l16;
      // B[kk][j] = K[j][kk]: consecutive kk contiguous in the K row
      v2f bf = *(const v2f*)(Kp + (size_t)j * 1536 + d + 2 * half);
      acc[t] = wmma4(a, bf, acc[t]);
    }
  }

  // stage scaled scores: row = half*8 + i, col = j0w + t*16 + l16
#pragma unroll
  for (int t = 0; t < 8; ++t)
#pragma unroll
    for (int i = 0; i < 8; ++i)
      S[(half * 8 + i) * 1024 + j0w + t * 16 + l16] = acc[t][i] * 0.125f;

  __syncthreads();

  // softmax: wave w handles rows 2w, 2w+1; 32 elements per lane
  float* __restrict__ outRow = attn + ((size_t)bh * 1024 + i0) * 1024;
  for (int rr = 0; rr < 2; ++rr) {
    const int row = wave * 2 + rr;
    float v[32];
    float m = -3.4e38f;
#pragma unroll
    for (int t = 0; t < 32; ++t) {
      v[t] = S[row * 1024 + lane + 32 * t];
      m = fmaxf(m, v[t]);
    }
#pragma unroll
    for (int off = 16; off > 0; off >>= 1)
      m = fmaxf(m, __shfl_xor(m, off, 32));
    float s = 0.f;
#pragma unroll
    for (int t = 0; t < 32; ++t) {
      v[t] = __expf(v[t] - m);
      s += v[t];
    }
#pragma unroll
    for (int off = 16; off > 0; off >>= 1)
      s += __shfl_xor(s, off, 32);
    const float inv = 1.0f / s;
    float* __restrict__ dst = outRow + (size_t)row * 1024;
#pragma unroll
    for (int t = 0; t < 32; ++t)
      dst[lane + 32 * t] = v[t] * inv;
  }
}

// ---------------------------------------------------------------------------
// In-place cross-head mix + LayerNorm over the head axis.
// attn[k] <- LN_h( sum_h attn[h] * rw[h,k] ) * g[k] + b[k]
// ---------------------------------------------------------------------------
__global__ __launch_bounds__(256) void reattn_mix_ln(
    float* __restrict__ attn, const float* __restrict__ rw,
    const float* __restrict__ g, const float* __restrict__ bb)
{
  __shared__ float srw[64], sg[8], sb[8];
  if (threadIdx.x < 64) srw[threadIdx.x] = rw[threadIdx.x];
  if (threadIdx.x < 8) { sg[threadIdx.x] = g[threadIdx.x]; sb[threadIdx.x] = bb[threadIdx.x]; }
  __syncthreads();

  const size_t nn  = 1024ull * 1024ull;
  const size_t p   = (size_t)blockIdx.x * blockDim.x + threadIdx.x;  // < 8*nn
  const size_t b   = p / nn;
  const size_t rem = p - b * nn;
  float* __restrict__ base = attn + b * 8 * nn + rem;

  float a[8];
#pragma unroll
  for (int h = 0; h < 8; ++h) a[h] = base[h * nn];

  float m[8];
#pragma unroll
  for (int k = 0; k < 8; ++k) {
    float acc = 0.f;
#pragma unroll
    for (int h = 0; h < 8; ++h) acc = fmaf(a[h], srw[h * 8 + k], acc);
    m[k] = acc;
  }

  float mu = 0.f;
#pragma unroll
  for (int k = 0; k < 8; ++k) mu += m[k];
  mu *= 0.125f;
  float var = 0.f;
#pragma unroll
  for (int k = 0; k < 8; ++k) { const float d = m[k] - mu; var = fmaf(d, d, var); }
  var *= 0.125f;
  const float rs = rsqrtf(var + 1e-5f);

#pragma unroll
  for (int k = 0; k < 8; ++k)
    base[k * nn] = (m[k] - mu) * rs * sg[k] + sb[k];
}

// ---------------------------------------------------------------------------
// O[b, i, h*64+d] = attn[b,h,i,:] @ V[b,:,h,d]
// block = 256 threads (8 waves); grid.x = bh (64), grid.y = 8.
// Wave owns 16 rows x 64 cols; shared V panel async-staged in LDS
// (64-deep K chunks, row stride padded to 80 words -> conflict-free).
// ---------------------------------------------------------------------------
__global__ __launch_bounds__(256) void attn_v(
    const float* __restrict__ attn, const float* __restrict__ qkv,
    float* __restrict__ O)
{
  __shared__ float Vs[64 * 80];    // [j within chunk][d], padded stride 80

  const int lane = threadIdx.x & 31;
  const int wave = threadIdx.x >> 5;
  const int half = lane >> 4;
  const int l16  = lane & 15;

  const int bh = blockIdx.x;
  const int b  = bh >> 3;
  const int h  = bh & 7;
  const int i0 = (blockIdx.y * 8 + wave) * 16;

  const float* __restrict__ Arow = attn + ((size_t)bh * 1024 + i0 + l16) * 1024;
  const float* __restrict__ V    = qkv + (size_t)(b * 1024) * 1536 + 1024 + h * 64;

  v8f acc[4] = {};
  for (int kc = 0; kc < 1024; kc += 64) {
    __syncthreads();
    // stage V[kc..kc+63][0..63]: 4096 floats = 1024 b128, 4 per thread
#pragma unroll
    for (int it = 0; it < 4; ++it) {
      const int idx = threadIdx.x + it * 256;   // 0..1023
      const int kk  = idx >> 4;                 // 0..63
      const int cc  = (idx & 15) * 4;           // 0..60
      async_ld_lds_b128(&Vs[kk * 80 + cc],
                        V + (size_t)(kc + kk) * 1536 + cc);
    }
    wait_async0();
    __syncthreads();

#pragma unroll
    for (int k4 = 0; k4 < 64; k4 += 4) {
      v2f a = *(const v2f*)(Arow + kc + k4 + 2 * half);
#pragma unroll
      for (int t = 0; t < 4; ++t) {
        v2f bf;
        bf.x = Vs[(k4 + 2 * half)     * 80 + t * 16 + l16];
        bf.y = Vs[(k4 + 2 * half + 1) * 80 + t * 16 + l16];
        acc[t] = wmma4(a, bf, acc[t]);
      }
    }
  }

#pragma unroll
  for (int t = 0; t < 4; ++t)
#pragma unroll
    for (int i = 0; i < 8; ++i)
      O[((size_t)b * 1024 + i0 + half * 8 + i) * 512 + h * 64 + t * 16 + l16] = acc[t][i];
}

// ---------------------------------------------------------------------------
extern "C" void kernel_launch(void* const* d_in, const int* in_sizes, int n_in,
                              void* d_out, int out_size, void* d_ws, size_t ws_size,
                              hipStream_t stream) {
  (void)in_sizes; (void)n_in; (void)out_size; (void)ws_size;

  const float* x        = (const float*)d_in[0];  // (8,1024,512)
  const float* W_qkv    = (const float*)d_in[1];  // (512,1536)
  const float* b_qkv    = (const float*)d_in[2];  // (1536,)
  const float* reattn_w = (const float*)d_in[3];  // (8,8)
  const float* ln_g     = (const float*)d_in[4];  // (8,)
  const float* ln_b     = (const float*)d_in[5];  // (8,)
  const float* W_out    = (const float*)d_in[6];  // (512,512)
  const float* b_out    = (const float*)d_in[7];  // (512,)

  float* out  = (float*)d_out;                    // (8,1024,512)
  float* attn = out + 8ull * 1024 * 512;          // (8,8,1024,1024)

  float* qkv = (float*)d_ws;                      // (8192,1536)
  float* O   = qkv + 8192ull * 1536;              // (8,1024,512)

  // 1) qkv = x @ W_qkv + b_qkv        (M=8192, N=1536, K=512)
  wmma_gemm_bias<<<dim3(1536 / 16, 8192 / 128), 256, 0, stream>>>(
      x, W_qkv, b_qkv, qkv, /*K=*/512, /*lda=*/512, /*ldb=*/1536, /*ldc=*/1536);

  // 2) attn = softmax(Q K^T * scale)  -> written into d_out attn region
  attn_scores_softmax<<<dim3(64, 64), 256, 0, stream>>>(qkv, attn);

  // 3) in-place cross-head mix + LayerNorm over heads
  reattn_mix_ln<<<dim3((8u * 1024 * 1024) / 256), 256, 0, stream>>>(
      attn, reattn_w, ln_g, ln_b);

  // 4) O = attn @ V   (written as (b, n, h*64+d) for contiguous out-proj)
  attn_v<<<dim3(64, 8), 256, 0, stream>>>(attn, qkv, O);

  // 5) out = O @ W_out + b_out        (M=8192, N=512, K=512)
  wmma_gemm_bias<<<dim3(512 / 16, 8192 / 128), 256, 0, stream>>>(
      O, W_out, b_out, out, /*K=*/512, /*lda=*/512, /*ldb=*/512, /*ldc=*/512);
}